// QuantumLayer_71133248356385
// MI455X (gfx1250) — compile-verified
//
#include <hip/hip_runtime.h>
#include <math.h>

#define DIMQ  65536   // 2^16 amplitudes
#define BATCH 16
#define NLAYERS 3

typedef __attribute__((ext_vector_type(8))) float v8f;
typedef __attribute__((ext_vector_type(2))) float v2f;

__device__ __forceinline__ float2 cmul(float2 a, float2 b) {
  return make_float2(a.x * b.x - a.y * b.y, a.x * b.y + a.y * b.x);
}

// ---------------------------------------------------------------------------
// Kernel 1: initial encoding layer. RZ(x)·RY(x)|0> per qubit -> product state.
// v_q[0] = (c^2, -c*s), v_q[1] = (s*c, s^2) with c=cos(x/2), s=sin(x/2).
// state[b][m] = prod over the 4 nibbles of per-nibble 16-entry tables.
// One block per batch element.
// ---------------------------------------------------------------------------
__global__ void __launch_bounds__(256) qc_init(const float* __restrict__ x,
                                               float2* __restrict__ out) {
  __shared__ float2 w[4][16];
  const int b = blockIdx.x;
  const int tid = threadIdx.x;
  if (tid < 64) {
    const int k = tid >> 4;          // nibble index (bits 4k..4k+3)
    const int j = tid & 15;          // nibble value
    float2 p = make_float2(1.f, 0.f);
#pragma unroll
    for (int t = 0; t < 4; ++t) {
      const int q = 15 - 4 * k - t;  // qubit owning bit (4k+t)
      const float h = 0.5f * x[b * 16 + q];
      const float ch = cosf(h), sh = sinf(h);
      float2 v;
      if ((j >> t) & 1) v = make_float2(sh * ch, sh * sh);
      else              v = make_float2(ch * ch, -ch * sh);
      p = cmul(p, v);
    }
    w[k][j] = p;
  }
  __syncthreads();
  float2* ob = out + (size_t)b * DIMQ;
  for (int m = tid; m < DIMQ; m += 256) {
    float2 r = cmul(cmul(w[3][(m >> 12) & 15], w[2][(m >> 8) & 15]),
                    cmul(w[1][(m >> 4) & 15], w[0][m & 15]));
    ob[m] = r;
  }
}

// ---------------------------------------------------------------------------
// Kernel 2: apply one fused 16x16 group unitary (kron of 4 RZ*RY) to nibble k.
// If `gather`, reads apply the previous layer's full CNOT chain via the
// Gray-code map idx ^= idx>>1 (state_after[m] = state_before[m ^ (m>>1)]).
// Complex 16x16 x 16x16-tile matmul via V_WMMA_F32_16X16X4_F32 (4 K-chunks,
// 4 real wmmas per chunk -> 16 wmma per wave-tile).
// grid = (BATCH, 32), block = 256 (8 waves); one 16-column tile per wave.
// ---------------------------------------------------------------------------
__global__ void __launch_bounds__(256) qc_apply(const float2* __restrict__ in,
                                                float2* __restrict__ out,
                                                const float* __restrict__ wts,
                                                int layer, int k, int gather) {
  __shared__ float sUr[16][16];
  __shared__ float sUi[16][16];
  const int tid = threadIdx.x;

  // Build fused unitary U = kron_t M_{q_t},  M = RZ(hz)*RY(hy).
  {
    const int jp = tid >> 4;   // row j'
    const int j  = tid & 15;   // col j
    float ur = 1.f, ui = 0.f;
#pragma unroll
    for (int t = 0; t < 4; ++t) {
      const int q = 15 - 4 * k - t;
      const float hy = 0.5f * wts[layer * 32 + q];
      const float hz = 0.5f * wts[layer * 32 + q + 16];
      const float cy = cosf(hy), sy = sinf(hy);
      const float cz = cosf(hz), sz = sinf(hz);
      const int rb = (jp >> t) & 1, cb = (j >> t) & 1;
      // RY entry (real): [[c,-s],[s,c]]
      const float e = (rb == cb) ? cy : (rb ? sy : -sy);
      // RZ phase on row rb: e^{-i hz} (row0) / e^{+i hz} (row1)
      const float mr = cz * e;
      const float mi = (rb ? sz : -sz) * e;
      const float nr = ur * mr - ui * mi;
      const float ni = ur * mi + ui * mr;
      ur = nr; ui = ni;
    }
    sUr[jp][j] = ur;
    sUi[jp][j] = ui;
  }
  __syncthreads();

  const int lane = tid & 31;
  const int wv   = tid >> 5;
  const int half = lane >> 4;        // lane half selects K/M sub-range
  const int ln   = lane & 15;        // M for A, N for B/C/D
  const int b    = blockIdx.x;
  const int tile = blockIdx.y * 8 + wv;   // 0..255
  const int shft = 4 * k;
  const int S    = 1 << shft;             // nibble stride
  const int col  = tile * 16 + ln;        // 0..4095, lane's column
  const int a    = col >> shft;
  const int c    = col & (S - 1);
  const int base_m = (a << (shft + 4)) | c;   // index with nibble j = 0
  const float2* ib = in + (size_t)b * DIMQ;
  float2* ob = out + (size_t)b * DIMQ;

  // A = U (M=ln, K=kk), B = X tile (K=kk, N=ln). Same K decomposition:
  // K = q*4 + vgpr + 2*half.   An = -Ui (f32 wmma has no A-negate).
  v2f Ar[4], Ai[4], An[4], Br[4], Bi[4];
#pragma unroll
  for (int q = 0; q < 4; ++q) {
#pragma unroll
    for (int v = 0; v < 2; ++v) {
      const int kk = q * 4 + v + 2 * half;
      const float r = sUr[ln][kk];
      const float i = sUi[ln][kk];
      Ar[q][v] = r;
      Ai[q][v] = i;
      An[q][v] = -i;
      int idx = base_m + (kk << shft);     // logical input index (j = kk)
      if (gather) idx ^= (idx >> 1);       // fold previous CNOT chain (Gray map)
      const float2 xv = ib[idx];
      Br[q][v] = xv.x;
      Bi[q][v] = xv.y;
    }
  }

  v8f accr = {0.f, 0.f, 0.f, 0.f, 0.f, 0.f, 0.f, 0.f};
  v8f acci = {0.f, 0.f, 0.f, 0.f, 0.f, 0.f, 0.f, 0.f};
#pragma unroll
  for (int q = 0; q < 4; ++q) {
    // Yr += Ur*Xr - Ui*Xi ; Yi += Ur*Xi + Ui*Xr
    accr = __builtin_amdgcn_wmma_f32_16x16x4_f32(false, Ar[q], false, Br[q],
                                                 (short)0, accr, false, false);
    accr = __builtin_amdgcn_wmma_f32_16x16x4_f32(false, An[q], false, Bi[q],
                                                 (short)0, accr, false, false);
    acci = __builtin_amdgcn_wmma_f32_16x16x4_f32(false, Ar[q], false, Bi[q],
                                                 (short)0, acci, false, false);
    acci = __builtin_amdgcn_wmma_f32_16x16x4_f32(false, Ai[q], false, Br[q],
                                                 (short)0, acci, false, false);
  }

  // D layout: VGPR r -> M = r + 8*half, N = ln. Store to logical index.
#pragma unroll
  for (int r = 0; r < 8; ++r) {
    const int M = r + 8 * half;
    ob[base_m + (M << shft)] = make_float2(accr[r], acci[r]);
  }
}

// ---------------------------------------------------------------------------
// Kernel 3: measurement. Applies final CNOT chain via Gray gather, then
// <Z_q> = sum_m (1 - 2*bit_{15-q}(m)) * |amp|^2.  One block per batch elem.
// ---------------------------------------------------------------------------
__global__ void __launch_bounds__(256) qc_measure(const float2* __restrict__ in,
                                                  float* __restrict__ out) {
  __shared__ float red[256];
  const int b = blockIdx.x;
  const int tid = threadIdx.x;
  float acc[16];
#pragma unroll
  for (int q = 0; q < 16; ++q) acc[q] = 0.f;

  const float2* ib = in + (size_t)b * DIMQ;
  for (int m = tid; m < DIMQ; m += 256) {
    const int p = m ^ (m >> 1);           // final layer's CNOT chain
    const float2 av = ib[p];
    const float pr = av.x * av.x + av.y * av.y;
#pragma unroll
    for (int q = 0; q < 16; ++q) {
      acc[q] += ((m >> (15 - q)) & 1) ? -pr : pr;
    }
  }
#pragma unroll 1
  for (int q = 0; q < 16; ++q) {
    red[tid] = acc[q];
    __syncthreads();
    for (int off = 128; off > 0; off >>= 1) {
      if (tid < off) red[tid] += red[tid + off];
      __syncthreads();
    }
    if (tid == 0) out[b * 16 + q] = red[0];
    __syncthreads();
  }
}

// ---------------------------------------------------------------------------
// Host launcher: 1 init + 12 fused-unitary WMMA passes (ping-pong in ws,
// 8 MB per buffer -> fully L2-resident) + 1 measurement pass.
// ---------------------------------------------------------------------------
extern "C" void kernel_launch(void* const* d_in, const int* in_sizes, int n_in,
                              void* d_out, int out_size, void* d_ws, size_t ws_size,
                              hipStream_t stream) {
  (void)in_sizes; (void)n_in; (void)out_size; (void)ws_size;
  const float* x   = (const float*)d_in[0];   // (16, 16) f32
  const float* wts = (const float*)d_in[1];   // (96,)   f32

  float2* bufA = (float2*)d_ws;
  float2* bufB = bufA + (size_t)BATCH * DIMQ;  // needs 2*8MB of workspace

  qc_init<<<dim3(BATCH), dim3(256), 0, stream>>>(x, bufA);

  float2* cur = bufA;
  float2* nxt = bufB;
  for (int l = 0; l < NLAYERS; ++l) {
    for (int p = 0; p < 4; ++p) {
      const int k = 3 - p;                       // nibble 3 = qubits 0..3 first
      const int gather = (p == 0 && l > 0) ? 1 : 0;  // fold prev layer's CNOTs
      qc_apply<<<dim3(BATCH, 32), dim3(256), 0, stream>>>(cur, nxt, wts, l, k,
                                                          gather);
      float2* t = cur; cur = nxt; nxt = t;
    }
  }

  qc_measure<<<dim3(BATCH), dim3(256), 0, stream>>>(cur, (float*)d_out);
}